// RecurrentGCN_7301444403385
// MI455X (gfx1250) — compile-verified
//
#include <hip/hip_runtime.h>
#include <math.h>

// ---------------------------------------------------------------------------
// DCRNN (DConv-GRU) cell for MI455X / gfx1250.
// N=50000 nodes, E=800000 edges, F=16, HID=32, C=F+HID=48, K=2.
//  - Edge diffusion: 16-lane-per-edge coalesced gather + f32 atomic scatter
//  - Dense GEMMs: V_WMMA_F32_16X16X4_F32 (fp32-exact matrix path),
//    A tiles staged global->LDS with GLOBAL_LOAD_ASYNC_TO_LDS_B128 (ASYNCcnt)
// ---------------------------------------------------------------------------

typedef float v2f __attribute__((ext_vector_type(2)));
typedef float v8f __attribute__((ext_vector_type(8)));

#define NF 16           // node feature dim
#define NH 32           // hidden dim
#define NC 48           // F + HID
#define NK 96           // concatenated K dim: [XH | Tx1]
#define LDS_STRIDE 100  // A-tile row stride in floats (row base 400B, 16B aligned)

__device__ __forceinline__ float sigmoidf_(float v) {
    return 1.0f / (1.0f + __expf(-v));
}

// Generic pointer -> 32-bit LDS byte offset (addrspace(3)).
__device__ __forceinline__ unsigned lds_offset_of(const void* p) {
    return (unsigned)(unsigned long long)
        (__attribute__((address_space(3))) const char*)p;
}

// Async direct-to-LDS 16B copy (GLOBAL_LOAD_ASYNC_TO_LDS_B128, GV mode).
__device__ __forceinline__ void async_copy_b128(unsigned lds_byte_off,
                                                const void* gaddr) {
    asm volatile("global_load_async_to_lds_b128 %0, %1, off"
                 :
                 : "v"(lds_byte_off), "v"(gaddr)
                 : "memory");
}

__device__ __forceinline__ void wait_async_zero() {
    asm volatile("s_wait_asynccnt 0x0" ::: "memory");
}

// ---------------------------------------------------------------------------
// Pre-sum diffusion weights: Wcat[g][r][h], r<48 -> W[0,0]+W[1,0] (identity
// term), r>=48 -> W[0,1]+W[1,1] (diffusion term).  W is [2,2,48,32].
// ---------------------------------------------------------------------------
__global__ void prep_wcat(const float* __restrict__ Wz,
                          const float* __restrict__ Wr,
                          const float* __restrict__ Wh,
                          float* __restrict__ wcat) {
    int gid = blockIdx.x * blockDim.x + threadIdx.x;
    if (gid >= 3 * NK * NH) return;
    int g   = gid / (NK * NH);
    int rem = gid % (NK * NH);
    int r = rem / NH;
    int h = rem % NH;
    int k = (r >= NC) ? 1 : 0;
    int c = (r >= NC) ? (r - NC) : r;
    const float* W = (g == 0) ? Wz : ((g == 1) ? Wr : Wh);
    size_t base = (size_t)c * NH + h;
    wcat[gid] = W[(size_t)k * NC * NH + base] +
                W[(size_t)(2 + k) * NC * NH + base];
}

// ---------------------------------------------------------------------------
// Bidirectional degree-normalized diffusion scatter, 16 lanes per edge.
// Lane L of an edge group owns columns {L, L+16, L+32} of the 48-wide row:
//   Tx1[dst] += (w/deg0[src]) * XH[src];   Tx1[src] += (w/deg1[src]) * XH[dst]
// XH = [x | hid] (hid = prev_hidden in pass 1, R*prev_hidden in pass 2).
// Gathers are 16 consecutive lane addresses -> coalesced; scatter-adds hit
// ~3 cachelines of global_atomic_add_f32 per direction.
// ---------------------------------------------------------------------------
__global__ void scatter_tx1(const int* __restrict__ edge,
                            const float* __restrict__ ew,
                            const float* __restrict__ deg,
                            const float* __restrict__ x,
                            const float* __restrict__ hid,
                            float* __restrict__ tx1, int N, int E) {
    int gid = blockIdx.x * blockDim.x + threadIdx.x;
    int e = gid >> 4;
    int L = gid & 15;
    if (e >= E) return;
    if (e + 4096 < E) __builtin_prefetch(&ew[e + 4096], 0, 1);

    int s = edge[e];
    int d = edge[E + e];
    float w  = ew[e];
    float no = w / deg[s];      // deg[0][src]
    float ni = w / deg[N + s];  // deg[1][src] (reference indexes both by src)

    float xs  = x[(size_t)s * NF + L];
    float hs0 = hid[(size_t)s * NH + L];
    float hs1 = hid[(size_t)s * NH + L + 16];
    float xd  = x[(size_t)d * NF + L];
    float hd0 = hid[(size_t)d * NH + L];
    float hd1 = hid[(size_t)d * NH + L + 16];

    float* td = tx1 + (size_t)d * NC;
    float* ts = tx1 + (size_t)s * NC;
    __hip_atomic_fetch_add(&td[L],      no * xs,  __ATOMIC_RELAXED, __HIP_MEMORY_SCOPE_AGENT);
    __hip_atomic_fetch_add(&td[L + 16], no * hs0, __ATOMIC_RELAXED, __HIP_MEMORY_SCOPE_AGENT);
    __hip_atomic_fetch_add(&td[L + 32], no * hs1, __ATOMIC_RELAXED, __HIP_MEMORY_SCOPE_AGENT);
    __hip_atomic_fetch_add(&ts[L],      ni * xd,  __ATOMIC_RELAXED, __HIP_MEMORY_SCOPE_AGENT);
    __hip_atomic_fetch_add(&ts[L + 16], ni * hd0, __ATOMIC_RELAXED, __HIP_MEMORY_SCOPE_AGENT);
    __hip_atomic_fetch_add(&ts[L + 32], ni * hd1, __ATOMIC_RELAXED, __HIP_MEMORY_SCOPE_AGENT);
}

// ---------------------------------------------------------------------------
// WMMA fragment helpers (V_WMMA_F32_16X16X4_F32 layouts, wave32):
//   A 16x4: lanes 0-15 hold M=lane, VGPR0/1 = K {0,1}; lanes 16-31 K {2,3}.
//   B  4x16: rows striped across lanes within a VGPR (mirrored split).
//   C/D 16x16: VGPR i -> M=i (lanes 0-15) / M=i+8 (lanes 16-31), N=lane&15.
// ---------------------------------------------------------------------------
__device__ __forceinline__ v2f fragA(const float* lds, int lane, int k0) {
    int m  = lane & 15;
    int kb = k0 + ((lane >> 4) << 1);
    v2f a;
    a.x = lds[m * LDS_STRIDE + kb];
    a.y = lds[m * LDS_STRIDE + kb + 1];
    return a;
}

__device__ __forceinline__ v2f fragB(const float* __restrict__ W /*[96][32]*/,
                                     int lane, int k0, int n0) {
    int n  = lane & 15;
    int kr = k0 + ((lane >> 4) << 1);
    v2f b;
    b.x = W[kr * NH + n0 + n];
    b.y = W[(kr + 1) * NH + n0 + n];
    return b;
}

// Stage one wave's A tile [16 x 96] = [x | hid | tx1] into LDS using
// async direct-to-LDS b128 copies: 384 16B chunks, 12 per lane.
__device__ __forceinline__ void load_tileA_async(float* lds,
                                                 const float* __restrict__ x,
                                                 const float* __restrict__ hid,
                                                 const float* __restrict__ tx1,
                                                 int tile, int lane) {
    unsigned base = lds_offset_of(lds);
#pragma unroll
    for (int i = 0; i < 12; ++i) {
        int id = lane + 32 * i;        // 0..383
        int r  = id / 24;              // tile row 0..15
        int j  = id % 24;              // 16B chunk 0..23 (cols j*4..j*4+3)
        int row = tile + r;
        const float* src;
        if (j < 4)       src = x   + (size_t)row * NF + j * 4;
        else if (j < 12) src = hid + (size_t)row * NH + (j - 4) * 4;
        else             src = tx1 + (size_t)row * NC + (j - 12) * 4;
        unsigned off = base + (unsigned)(r * LDS_STRIDE + j * 4) * 4u;
        async_copy_b128(off, src);
    }
    wait_async_zero();
}

// ---------------------------------------------------------------------------
// Pass-1 GEMM: Z = sigmoid([XH|Tx1] @ Wzc + bz), R likewise; store Z, R*prev.
// One wave per 16-row tile, 4 waves per block.
// ---------------------------------------------------------------------------
__global__ void __launch_bounds__(128)
gemm_zr(const float* __restrict__ x, const float* __restrict__ prev,
        const float* __restrict__ tx1, const float* __restrict__ Wzc,
        const float* __restrict__ Wrc, const float* __restrict__ bz,
        const float* __restrict__ br, float* __restrict__ Z,
        float* __restrict__ Rh, int N) {
    __shared__ float ldsA[4][16 * LDS_STRIDE];
    int wave = threadIdx.x >> 5;
    int lane = threadIdx.x & 31;
    int tile = (blockIdx.x * 4 + wave) * 16;
    if (tile + 16 > N) tile = N - 16;  // clamp; duplicate tiles write same data
    float* lds = ldsA[wave];
    load_tileA_async(lds, x, prev, tx1, tile, lane);
    __syncthreads();

    v8f accz[2] = {{0,0,0,0,0,0,0,0}, {0,0,0,0,0,0,0,0}};
    v8f accr[2] = {{0,0,0,0,0,0,0,0}, {0,0,0,0,0,0,0,0}};
    for (int k0 = 0; k0 < NK; k0 += 4) {
        v2f a = fragA(lds, lane, k0);
#pragma unroll
        for (int t = 0; t < 2; ++t) {
            int n0 = t * 16;
            v2f bzf = fragB(Wzc, lane, k0, n0);
            accz[t] = __builtin_amdgcn_wmma_f32_16x16x4_f32(
                false, a, false, bzf, (short)0, accz[t], false, false);
            v2f brf = fragB(Wrc, lane, k0, n0);
            accr[t] = __builtin_amdgcn_wmma_f32_16x16x4_f32(
                false, a, false, brf, (short)0, accr[t], false, false);
        }
    }

    int half  = lane >> 4;
    int nlane = lane & 15;
#pragma unroll
    for (int t = 0; t < 2; ++t) {
        int col = t * 16 + nlane;
        float bzv = bz[col], brv = br[col];
#pragma unroll
        for (int i = 0; i < 8; ++i) {
            int row = tile + i + half * 8;
            float z = sigmoidf_(accz[t][i] + bzv);
            float r = sigmoidf_(accr[t][i] + brv);
            Z[(size_t)row * NH + col]  = z;
            Rh[(size_t)row * NH + col] = r * prev[(size_t)row * NH + col];
        }
    }
}

// ---------------------------------------------------------------------------
// Pass-2 GEMM: H~ = tanh([x|R*prev|Tx1b] @ Whc + bh);
// Hn = Z*prev + (1-Z)*H~ ; store Hn.
// ---------------------------------------------------------------------------
__global__ void __launch_bounds__(128)
gemm_h(const float* __restrict__ x, const float* __restrict__ prev,
       const float* __restrict__ Rh, const float* __restrict__ tx1,
       const float* __restrict__ Whc, const float* __restrict__ bh,
       const float* __restrict__ Z, float* __restrict__ Hn, int N) {
    __shared__ float ldsA[4][16 * LDS_STRIDE];
    int wave = threadIdx.x >> 5;
    int lane = threadIdx.x & 31;
    int tile = (blockIdx.x * 4 + wave) * 16;
    if (tile + 16 > N) tile = N - 16;
    float* lds = ldsA[wave];
    load_tileA_async(lds, x, Rh, tx1, tile, lane);
    __syncthreads();

    v8f acch[2] = {{0,0,0,0,0,0,0,0}, {0,0,0,0,0,0,0,0}};
    for (int k0 = 0; k0 < NK; k0 += 4) {
        v2f a = fragA(lds, lane, k0);
#pragma unroll
        for (int t = 0; t < 2; ++t) {
            v2f bf = fragB(Whc, lane, k0, t * 16);
            acch[t] = __builtin_amdgcn_wmma_f32_16x16x4_f32(
                false, a, false, bf, (short)0, acch[t], false, false);
        }
    }

    int half  = lane >> 4;
    int nlane = lane & 15;
#pragma unroll
    for (int t = 0; t < 2; ++t) {
        int col = t * 16 + nlane;
        float bhv = bh[col];
#pragma unroll
        for (int i = 0; i < 8; ++i) {
            int row = tile + i + half * 8;
            float ht = tanhf(acch[t][i] + bhv);
            float z  = Z[(size_t)row * NH + col];
            float pv = prev[(size_t)row * NH + col];
            Hn[(size_t)row * NH + col] = z * pv + (1.0f - z) * ht;
        }
    }
}

// out[n] = relu(Hn[n]) . lin_w + lin_b
__global__ void final_lin(const float* __restrict__ Hn,
                          const float* __restrict__ lw,
                          const float* __restrict__ lb,
                          float* __restrict__ out, int N) {
    int n = blockIdx.x * blockDim.x + threadIdx.x;
    if (n >= N) return;
    float acc = 0.0f;
#pragma unroll
    for (int h = 0; h < NH; ++h) {
        float v = Hn[(size_t)n * NH + h];
        v = v > 0.0f ? v : 0.0f;
        acc += v * lw[h];
    }
    out[n] = acc + lb[0];
}

// ---------------------------------------------------------------------------
extern "C" void kernel_launch(void* const* d_in, const int* in_sizes, int n_in,
                              void* d_out, int out_size, void* d_ws, size_t ws_size,
                              hipStream_t stream) {
    const float* x    = (const float*)d_in[0];
    const int*   edge = (const int*)d_in[1];
    const float* ew   = (const float*)d_in[2];
    const float* deg  = (const float*)d_in[3];
    const float* prev = (const float*)d_in[4];
    const float* Wz   = (const float*)d_in[5];
    const float* bz   = (const float*)d_in[6];
    const float* Wr   = (const float*)d_in[7];
    const float* br   = (const float*)d_in[8];
    const float* Wh   = (const float*)d_in[9];
    const float* bh   = (const float*)d_in[10];
    const float* lw   = (const float*)d_in[11];
    const float* lb   = (const float*)d_in[12];
    float* out = (float*)d_out;

    int N = in_sizes[0] / NF;
    int E = in_sizes[2];

    float* ws   = (float*)d_ws;
    float* Tx1  = ws;                       // N*48
    float* Zb   = Tx1 + (size_t)N * NC;     // N*32
    float* Rh   = Zb  + (size_t)N * NH;     // N*32
    float* Hn   = Rh  + (size_t)N * NH;     // N*32
    float* Wcat = Hn  + (size_t)N * NH;     // 3*96*32
    float* Wzc = Wcat;
    float* Wrc = Wcat + NK * NH;
    float* Whc = Wcat + 2 * NK * NH;

    int nblk  = (N + 63) / 64;
    long long nscat = (long long)E * 16;
    int sblk  = (int)((nscat + 255) / 256);

    hipMemsetAsync(Tx1, 0, (size_t)N * NC * sizeof(float), stream);
    prep_wcat<<<(3 * NK * NH + 255) / 256, 256, 0, stream>>>(Wz, Wr, Wh, Wcat);
    scatter_tx1<<<sblk, 256, 0, stream>>>(edge, ew, deg, x, prev, Tx1, N, E);
    gemm_zr<<<nblk, 128, 0, stream>>>(x, prev, Tx1, Wzc, Wrc, bz, br, Zb, Rh, N);
    hipMemsetAsync(Tx1, 0, (size_t)N * NC * sizeof(float), stream);
    scatter_tx1<<<sblk, 256, 0, stream>>>(edge, ew, deg, x, Rh, Tx1, N, E);
    gemm_h<<<nblk, 128, 0, stream>>>(x, prev, Rh, Tx1, Whc, bh, Zb, Hn, N);
    final_lin<<<(N + 255) / 256, 256, 0, stream>>>(Hn, lw, lb, out, N);
}